// LML_1726576855866
// MI455X (gfx1250) — compile-verified
//
#include <hip/hip_runtime.h>

#define ROW_N   16384
#define NBRANCH 10
#define EPSV    1e-4f
#define NITER   100
#define TARGETF 1000.0f
#define NTHREADS 256
#define ELEMS_PER_THREAD (ROW_N / NTHREADS)   // 64
#define NWAVES (NTHREADS / 32)                // 8

typedef unsigned int v4u  __attribute__((ext_vector_type(4)));
typedef int          v8i_t __attribute__((ext_vector_type(8)));
typedef int          v4i_t __attribute__((ext_vector_type(4)));

#ifndef __has_builtin
#define __has_builtin(x) 0
#endif

#if __has_builtin(__builtin_amdgcn_tensor_load_to_lds) && __has_builtin(__builtin_amdgcn_s_wait_tensorcnt)
#define HAVE_TDM 1
#else
#define HAVE_TDM 0
#endif

#if __has_builtin(__builtin_amdgcn_tanhf)
#define USE_TANH 1
#else
#define USE_TANH 0
#endif

__device__ __forceinline__ float fast_exp2(float v) {
#if __has_builtin(__builtin_amdgcn_exp2f)
    return __builtin_amdgcn_exp2f(v);
#else
    return exp2f(v);
#endif
}

__device__ __forceinline__ float fast_rcp(float v) {
#if __has_builtin(__builtin_amdgcn_rcpf)
    return __builtin_amdgcn_rcpf(v);
#else
    return 1.0f / v;
#endif
}

// float -> key such that DESCENDING float order == ascending unsigned key order
__device__ __forceinline__ unsigned f2key(float f) {
    unsigned u = __float_as_uint(f);
    return (u & 0x80000000u) ? u : ((~u) & 0x7FFFFFFFu);
}
__device__ __forceinline__ float key2f(unsigned k) {
    unsigned u = (k & 0x80000000u) ? k : ((~k) & 0x7FFFFFFFu);
    return __uint_as_float(u);
}

// linspace(0,1,10)[i], exact endpoints like jnp.linspace
__device__ __forceinline__ float ls_of(int i) {
    return (i >= 9) ? 1.0f : (float)i * (1.0f / 9.0f);
}

// Find the element of rank `rank` (0-indexed, in descending order) among the
// ROW_N keys in LDS. 4 passes of 8-bit MSB-first radix select, 256 threads.
__device__ float block_radix_select(unsigned* keys, unsigned* hist,
                                    unsigned* s_prefix, unsigned* s_rank,
                                    int tid, unsigned rank) {
    unsigned prefix = 0, pmask = 0, rk = rank;
    for (int shift = 24; shift >= 0; shift -= 8) {
        hist[tid] = 0;
        __syncthreads();
        for (int i = tid; i < ROW_N; i += NTHREADS) {
            unsigned k = keys[i];
            if ((k & pmask) == prefix)
                atomicAdd(&hist[(k >> shift) & 255u], 1u);
        }
        __syncthreads();
        if (tid == 0) {
            unsigned c = 0;
            for (unsigned b = 0; b < 256; ++b) {
                unsigned hc = hist[b];
                if (c + hc > rk) { *s_prefix = prefix | (b << shift); *s_rank = rk - c; break; }
                c += hc;
            }
        }
        __syncthreads();
        prefix = *s_prefix;
        rk     = *s_rank;
        pmask |= (0xFFu << shift);
        __syncthreads();
    }
    return key2f(prefix);
}

__global__ __launch_bounds__(NTHREADS) void lml_kernel(const float* __restrict__ x,
                                                       float* __restrict__ y,
                                                       int nrows) {
    extern __shared__ unsigned char smem_raw[];
    unsigned* row   = (unsigned*)smem_raw;      // ROW_N keys/floats (64 KB)
    unsigned* hist  = row + ROW_N;              // 256
    unsigned* selp  = hist + 256;               // 1
    unsigned* selr  = selp + 1;                 // 1
    float*    wsum  = (float*)(selr + 1);       // NWAVES * NBRANCH

    const int tid  = threadIdx.x;
    const int r0   = blockIdx.x;
    if (r0 >= nrows) return;
    const float* xrow = x + (size_t)r0 * ROW_N;
    float*       yrow = y + (size_t)r0 * ROW_N;

    float h[ELEMS_PER_THREAD];   // pre-scaled row data, lives in VGPRs

#if HAVE_TDM
    // ---- Tensor Data Mover: DMA the whole 64KB row into LDS (wave 0 issues) ----
    if (tid < 32) {
        unsigned ldsOff = (unsigned)(unsigned long long)(void*)row; // flat LDS addr low 32b = LDS offset
        unsigned long long ga = (unsigned long long)xrow;
        v4u g0;
        g0.x = 1u;                                          // count=1, user descriptor
        g0.y = ldsOff;                                      // lds_addr
        g0.z = (unsigned)(ga & 0xFFFFFFFFull);              // global_addr[31:0]
        g0.w = (unsigned)((ga >> 32) & 0x01FFFFFFull) | (2u << 30); // addr[56:32] | type=2
        v8i_t g1;
        g1[0] = (int)(2u << 16);                            // data_size=4B, mask=0, no flags
        g1[1] = (int)((unsigned)ROW_N << 16);               // tensor_dim0[15:0]
        g1[2] = (int)(((unsigned)ROW_N >> 16) | (1u << 16));// dim0 hi | tensor_dim1=1
        g1[3] = (int)((unsigned)ROW_N << 16);               // dim1 hi=0 | tile_dim0=16384
        g1[4] = (int)1;                                     // tile_dim1=1, tile_dim2 unused
        g1[5] = (int)ROW_N;                                 // tensor_dim0_stride lo32
        g1[6] = (int)((unsigned)ROW_N << 16);               // stride0 hi=0 | stride1 lo16
        g1[7] = 0;                                          // stride1 hi
        v4i_t g2 = {0, 0, 0, 0};
        v4i_t g3 = {0, 0, 0, 0};
        v8i_t g4 = {0, 0, 0, 0, 0, 0, 0, 0};
        // clang-23 / therock-10.0 arity: (g0 v4u, g1 v8i, g2 v4i, g3 v4i, g4 v8i, cpol)
        __builtin_amdgcn_tensor_load_to_lds(g0, g1, g2, g3, g4, 0);
        __builtin_amdgcn_s_wait_tensorcnt(0);
    }
    __syncthreads();
    // read row from LDS into registers, overwrite LDS with sortable keys
#pragma unroll
    for (int j = 0; j < ELEMS_PER_THREAD; ++j) {
        int i = tid + j * NTHREADS;
        float f = __uint_as_float(row[i]);
#if USE_TANH
        h[j] = 0.5f * f;                         // sigmoid(t)=0.5+0.5*tanh(t/2)
#else
        h[j] = -1.4426950408889634f * f;         // exp(-t)=exp2(-t*log2e)
#endif
        row[i] = f2key(f);
    }
#else
    // fallback: stage keys from global directly
#pragma unroll
    for (int j = 0; j < ELEMS_PER_THREAD; ++j) {
        int i = tid + j * NTHREADS;
        float f = xrow[i];
#if USE_TANH
        h[j] = 0.5f * f;
#else
        h[j] = -1.4426950408889634f * f;
#endif
        row[i] = f2key(f);
    }
#endif
    __syncthreads();

    // ---- exact rank-999 / rank-1000 (descending) via radix select ----
    float vA = block_radix_select(row, hist, selp, selr, tid, 999u);   // x_sorted[N-1]
    float vB = block_radix_select(row, hist, selp, selr, tid, 1000u);  // x_sorted[N]

    float nl = -vA - 7.0f;   // nu_lower
    float nu = -vB + 7.0f;   // nu_upper

    const int wave = tid >> 5;
    const int lane = tid & 31;

    // ---- branched bisection (uniform across block) ----
    for (int it = 0; it < NITER; ++it) {
        float r = nu - nl;
        if (!(r > EPSV)) break;

        float cb[NBRANCH], part[NBRANCH];
#pragma unroll
        for (int b = 0; b < NBRANCH; ++b) {
            float nub = nl + r * ls_of(b);
#if USE_TANH
            cb[b] = 0.5f * nub;
#else
            cb[b] = -1.4426950408889634f * nub;
#endif
            part[b] = 0.0f;
        }

#pragma unroll
        for (int j = 0; j < ELEMS_PER_THREAD; ++j) {
#pragma unroll
            for (int b = 0; b < NBRANCH; ++b) {
#if USE_TANH
                part[b] += __builtin_amdgcn_tanhf(h[j] + cb[b]);
#else
                float e = fast_exp2(h[j] + cb[b]);
                part[b] += fast_rcp(1.0f + e);
#endif
            }
        }

        // wave32 reduction then cross-wave combine in LDS
#pragma unroll
        for (int b = 0; b < NBRANCH; ++b)
            for (int m = 16; m > 0; m >>= 1)
                part[b] += __shfl_xor(part[b], m, 32);
        if (lane == 0) {
#pragma unroll
            for (int b = 0; b < NBRANCH; ++b) wsum[wave * NBRANCH + b] = part[b];
        }
        __syncthreads();

        int cnt = 0;
#pragma unroll
        for (int b = 0; b < NBRANCH; ++b) {
            float s = 0.0f;
#pragma unroll
            for (int w = 0; w < NWAVES; ++w) s += wsum[w * NBRANCH + b];
#if USE_TANH
            float f = 0.5f * s + (0.5f * (float)ROW_N - TARGETF);
#else
            float f = s - TARGETF;
#endif
            cnt += (f < 0.0f) ? 1 : 0;
        }

        int  il = cnt - 1;
        bool J  = (il < 0);
        if (il < 0) il = 0;
        int iu = il + 1;
        if (iu > 9) iu = 9;                 // matches JAX take_along_axis clamping
        float nnl = nl + r * ls_of(il);
        float nnu = nl + r * ls_of(iu);
        if (J) nnl -= 7.0f;
        nl = nnl;
        nu = nnu;
        __syncthreads();                     // wsum reused next iteration
    }

    float nuf = nl + 0.5f * (nu - nl);

    // ---- final y = sigmoid(x + nu), straight from registers ----
#if USE_TANH
    float c = 0.5f * nuf;
#pragma unroll
    for (int j = 0; j < ELEMS_PER_THREAD; ++j)
        yrow[tid + j * NTHREADS] = 0.5f + 0.5f * __builtin_amdgcn_tanhf(h[j] + c);
#else
    float c = -1.4426950408889634f * nuf;
#pragma unroll
    for (int j = 0; j < ELEMS_PER_THREAD; ++j) {
        float e = fast_exp2(h[j] + c);
        yrow[tid + j * NTHREADS] = fast_rcp(1.0f + e);
    }
#endif
}

extern "C" void kernel_launch(void* const* d_in, const int* in_sizes, int n_in,
                              void* d_out, int out_size, void* d_ws, size_t ws_size,
                              hipStream_t stream) {
    (void)n_in; (void)out_size; (void)d_ws; (void)ws_size;
    const float* x = (const float*)d_in[0];
    float* yp = (float*)d_out;
    int rows = in_sizes[0] / ROW_N;
    size_t smem = (size_t)(ROW_N + 256 + 2) * sizeof(unsigned) +
                  (size_t)NWAVES * NBRANCH * sizeof(float);
    lml_kernel<<<dim3(rows), dim3(NTHREADS), smem, stream>>>(x, yp, rows);
}